// hireNet_60902636257970
// MI455X (gfx1250) — compile-verified
//
#include <hip/hip_runtime.h>
#include <math.h>

typedef __attribute__((ext_vector_type(2))) float v2f;
typedef __attribute__((ext_vector_type(8))) float v8f;

#define NPG   256
#define NGRAPH 256
#define NN    (NPG * NGRAPH)     // 65536 nodes
#define FD    128                // feature/hidden dim
#define EE    (NN * 16)          // 1,048,576 edges
#define NCLS  10

// ---------------------------------------------------------------------------
// small utility kernels
// ---------------------------------------------------------------------------
__global__ void k_fill(float* p, float v, int n) {
  int i = blockIdx.x * blockDim.x + threadIdx.x;
  if (i < n) p[i] = v;
}

// deg[dst] += emask   (emask is exactly 0.0 or 1.0 -> deterministic atomics)
__global__ void k_deg(const int* __restrict__ dst, const float* __restrict__ em,
                      float* __restrict__ deg) {
  int e = blockIdx.x * blockDim.x + threadIdx.x;
  float m = em[e];
  if (m != 0.f) atomicAdd(&deg[dst[e]], m);
}

__global__ void k_dinv(const float* __restrict__ deg, const float* __restrict__ nm,
                       float* __restrict__ dinv) {
  int n = blockIdx.x * blockDim.x + threadIdx.x;
  float d = deg[n] + nm[n];
  dinv[n] = (d > 0.f) ? (1.0f / sqrtf(d)) : 0.f;
}

// A[g][dstLocal][srcLocal] += emask * dinv[src] * dinv[dst]
// duplicate edges add identical values -> order-independent (deterministic)
__global__ void k_buildA(const int* __restrict__ src, const int* __restrict__ dst,
                         const float* __restrict__ em, const float* __restrict__ dinv,
                         float* __restrict__ A) {
  int e = blockIdx.x * blockDim.x + threadIdx.x;
  float m = em[e];
  int s = src[e], d = dst[e];
  float w = m * dinv[s] * dinv[d];
  if (w != 0.f) {
    size_t g = (size_t)(s >> 8);
    atomicAdd(&A[(g << 16) + (size_t)((d & 255) << 8) + (size_t)(s & 255)], w);
  }
}

// fold self-loop term into diagonal: A[g][i][i] += dinv^2 * nmask (runs after scatter)
__global__ void k_diag(float* __restrict__ A, const float* __restrict__ dinv,
                       const float* __restrict__ nm) {
  int n = blockIdx.x * blockDim.x + threadIdx.x;
  size_t g = (size_t)(n >> 8);
  int i = n & 255;
  float dv = dinv[n];
  A[(g << 16) + (size_t)(i << 8) + (size_t)i] += dv * dv * nm[n];
}

// ---------------------------------------------------------------------------
// FP32 WMMA GEMM:  C[128rows x 128] = A_blk[128 x K] @ B_blk[K x 128]
// block = 256 threads = 8 waves, each wave computes a 16x128 strip.
// B_blk staged in dynamic LDS (K*128 floats). bPeriod lets 2 blocks share one
// B operand (batched adjacency GEMM uses 2 blocks per graph).
// ---------------------------------------------------------------------------
__global__ void __launch_bounds__(256)
k_gemm_wmma(const float* __restrict__ A, const float* __restrict__ B,
            float* __restrict__ C, int K, int strideB, int bPeriod) {
  extern __shared__ float Blds[];
  int bid = blockIdx.x;
  const float* Ab = A + (size_t)bid * 128 * (size_t)K;
  const float* Bb = B + (size_t)(bid / bPeriod) * (size_t)strideB;
  float* Cb = C + (size_t)bid * 128 * 128;
  int tid = threadIdx.x;

  // cooperative load of B (K x 128, row-major) into LDS with float4
  {
    const float4* B4 = (const float4*)Bb;
    float4* L4 = (float4*)Blds;
    int tot4 = K * 32;
    for (int i = tid; i < tot4; i += 256) L4[i] = B4[i];
  }
  __syncthreads();

  int wave = tid >> 5, lane = tid & 31;
  int m = lane & 15;        // row within 16-row tile (A) / column (B,C)
  int hi = lane >> 4;       // lane group
  int rowBase = wave * 16;

  v8f c[8];
#pragma unroll
  for (int t = 0; t < 8; ++t) {
#pragma unroll
    for (int r = 0; r < 8; ++r) c[t][r] = 0.0f;
  }

  int ksteps = K >> 2;
  for (int ks = 0; ks < ksteps; ++ks) {
    int k0 = ks * 4 + 2 * hi;  // lanes 0-15 -> K=k,k+1 ; lanes 16-31 -> K=k+2,k+3
    // A fragment 16x4 f32 (ISA layout), 8B-aligned vector load
    v2f a = *(const v2f*)(Ab + (size_t)(rowBase + m) * (size_t)K + (size_t)k0);
#pragma unroll
    for (int t = 0; t < 8; ++t) {
      int n = t * 16 + m;
      v2f bf;
      bf.x = Blds[k0 * 128 + n];
      bf.y = Blds[k0 * 128 + 128 + n];
      c[t] = __builtin_amdgcn_wmma_f32_16x16x4_f32(
          /*neg_a=*/false, a, /*neg_b=*/false, bf,
          /*c_mod=*/(short)0, c[t], /*reuse_a=*/false, /*reuse_b=*/false);
    }
  }

  // C/D layout: VGPR r -> rows (rowBase + hi*8 + r), cols lane%16 within n-tile
#pragma unroll
  for (int t = 0; t < 8; ++t) {
#pragma unroll
    for (int r = 0; r < 8; ++r) {
      int row = rowBase + hi * 8 + r;
      int col = t * 16 + m;
      Cb[(size_t)row * 128 + col] = c[t][r];
    }
  }
}

// conv epilogue (in place): v = (agg + b) * nmask; relu
__global__ void k_combine(float* __restrict__ AG, const float* __restrict__ bias,
                          const float* __restrict__ nm) {
  int idx = blockIdx.x * 256 + threadIdx.x;
  int n = idx >> 7, f = idx & 127;
  float v = (AG[idx] + bias[f]) * nm[n];
  AG[idx] = v > 0.f ? v : 0.f;
}

// slin[n] = HR[n,:] . Wp   (one wave per node)
__global__ void k_slin(const float* __restrict__ H, const float* __restrict__ Wp,
                       float* __restrict__ slin) {
  int tid = threadIdx.x;
  int wave = tid >> 5, lane = tid & 31;
  int n = blockIdx.x * 8 + wave;
  const float* row = H + ((size_t)n << 7);
  float v = 0.f;
#pragma unroll
  for (int j = 0; j < 4; ++j) v += row[lane + 32 * j] * Wp[lane + 32 * j];
  for (int off = 16; off > 0; off >>= 1) v += __shfl_down(v, off, 32);
  if (lane == 0) slin[n] = v;
}

// s[n] = (A[g][i,:] . slin_g + bp) * nmask[n]   (one wave per node)
__global__ void k_score(const float* __restrict__ A, const float* __restrict__ slin,
                        const float* __restrict__ bp, const float* __restrict__ nm,
                        float* __restrict__ s) {
  int tid = threadIdx.x;
  int wave = tid >> 5, lane = tid & 31;
  int n = blockIdx.x * 8 + wave;
  size_t g = (size_t)(n >> 8);
  int i = n & 255;
  const float* Ar = A + (g << 16) + (size_t)(i << 8);
  const float* sl = slin + (g << 8);
  float v = 0.f;
#pragma unroll
  for (int j = 0; j < 8; ++j) v += Ar[lane + 32 * j] * sl[lane + 32 * j];
  for (int off = 16; off > 0; off >>= 1) v += __shfl_down(v, off, 32);
  if (lane == 0) s[n] = (v + bp[0]) * nm[n];
}

// exact per-graph top-k with JAX tie-break (value desc, index asc); nmask in place
__global__ void k_topk(const float* __restrict__ s, float* __restrict__ nm, int k) {
  __shared__ float sc[256];
  int g = blockIdx.x, t = threadIdx.x;
  int n = (g << 8) + t;
  float sv = (nm[n] > 0.f) ? s[n] : -INFINITY;
  sc[t] = sv;
  __syncthreads();
  int r = 0;
  for (int j = 0; j < 256; ++j) {
    float o = sc[j];
    r += (o > sv) || (o == sv && j < t);
  }
  nm[n] = (r < k) ? 1.f : 0.f;
}

// Xc = HR * tanh(s) * newmask
__global__ void k_pool(const float* __restrict__ HR, const float* __restrict__ s,
                       const float* __restrict__ nm, float* __restrict__ Xc) {
  int idx = blockIdx.x * 256 + threadIdx.x;
  int n = idx >> 7;
  Xc[idx] = HR[idx] * tanhf(s[n]) * nm[n];
}

__global__ void k_emask(const int* __restrict__ src, const int* __restrict__ dst,
                        const float* __restrict__ nm, float* __restrict__ em) {
  int e = blockIdx.x * blockDim.x + threadIdx.x;
  em[e] *= nm[src[e]] * nm[dst[e]];
}

// readout: z[g][0:128]=max over kept, z[g][128:256]=sum/cnt ; one block (128 thr)/graph
__global__ void k_readout(const float* __restrict__ X, const float* __restrict__ nm,
                          float* __restrict__ z, int acc) {
  int g = blockIdx.x, f = threadIdx.x;
  float mx = -INFINITY, sum = 0.f, cnt = 0.f;
  for (int i = 0; i < 256; ++i) {
    int n = (g << 8) + i;
    float m = nm[n];
    float v = X[((size_t)n << 7) + f];
    sum += v;
    cnt += m;
    if (m > 0.f && v > mx) mx = v;
  }
  float mean = sum / cnt;
  if (acc) {
    z[(g << 8) + f] += mx;
    z[(g << 8) + 128 + f] += mean;
  } else {
    z[(g << 8) + f] = mx;
    z[(g << 8) + 128 + f] = mean;
  }
}

// ---------------------------------------------------------------------------
// MLP head (tiny; per-graph blocks)
// ---------------------------------------------------------------------------
__global__ void k_mlp1(const float* __restrict__ z, const float* __restrict__ W,
                       const float* __restrict__ bias, float* __restrict__ z1) {
  __shared__ float zr[256];
  int g = blockIdx.x, t = threadIdx.x;  // 128 threads
  zr[t] = z[(g << 8) + t];
  zr[t + 128] = z[(g << 8) + 128 + t];
  __syncthreads();
  float acc = bias[t];
  for (int k = 0; k < 256; ++k) acc += zr[k] * W[k * 128 + t];
  z1[(g << 7) + t] = acc > 0.f ? acc : 0.f;
}

__global__ void k_mlp2(const float* __restrict__ z1, const float* __restrict__ W,
                       const float* __restrict__ bias, float* __restrict__ z2) {
  __shared__ float zr[128];
  int g = blockIdx.x, t = threadIdx.x;  // 64 threads
  zr[t] = z1[(g << 7) + t];
  zr[t + 64] = z1[(g << 7) + 64 + t];
  __syncthreads();
  float acc = bias[t];
  for (int k = 0; k < 128; ++k) acc += zr[k] * W[k * 64 + t];
  z2[(g << 6) + t] = acc > 0.f ? acc : 0.f;
}

__global__ void k_mlp3(const float* __restrict__ z2, const float* __restrict__ W,
                       const float* __restrict__ bias, float* __restrict__ out) {
  __shared__ float zr[64];
  __shared__ float lg[NCLS];
  int g = blockIdx.x, t = threadIdx.x;  // 32 threads (one wave)
  zr[t] = z2[(g << 6) + t];
  zr[t + 32] = z2[(g << 6) + 32 + t];
  __syncthreads();
  if (t < NCLS) {
    float acc = bias[t];
    for (int k = 0; k < 64; ++k) acc += zr[k] * W[k * NCLS + t];
    lg[t] = acc;
  }
  __syncthreads();
  if (t < NCLS) {
    float m = lg[0];
    for (int j = 1; j < NCLS; ++j) m = fmaxf(m, lg[j]);
    float se = 0.f;
    for (int j = 0; j < NCLS; ++j) se += expf(lg[j] - m);
    out[g * NCLS + t] = lg[t] - m - logf(se);
  }
}

// ---------------------------------------------------------------------------
extern "C" void kernel_launch(void* const* d_in, const int* in_sizes, int n_in,
                              void* d_out, int out_size, void* d_ws, size_t ws_size,
                              hipStream_t stream) {
  (void)in_sizes; (void)n_in; (void)out_size; (void)ws_size;

  const float* x = (const float*)d_in[0];
  const int* ei = (const int*)d_in[1];
  const int* src = ei;
  const int* dst = ei + EE;
  // d_in[2] = batch (implied by node layout; unused)
  const float* Wc[3] = {(const float*)d_in[3], (const float*)d_in[7], (const float*)d_in[11]};
  const float* bc[3] = {(const float*)d_in[4], (const float*)d_in[8], (const float*)d_in[12]};
  const float* Wp[3] = {(const float*)d_in[5], (const float*)d_in[9], (const float*)d_in[13]};
  const float* bp[3] = {(const float*)d_in[6], (const float*)d_in[10], (const float*)d_in[14]};
  const float* Wl1 = (const float*)d_in[15];
  const float* bl1 = (const float*)d_in[16];
  const float* Wl2 = (const float*)d_in[17];
  const float* bl2 = (const float*)d_in[18];
  const float* Wl3 = (const float*)d_in[19];
  const float* bl3 = (const float*)d_in[20];
  float* out = (float*)d_out;

  // workspace carve (256B aligned)
  char* ws = (char*)d_ws;
  size_t off = 0;
  auto carve = [&](size_t bytes) -> float* {
    float* p = (float*)(ws + off);
    off += (bytes + 255) & ~(size_t)255;
    return p;
  };
  float* Xc  = carve((size_t)NN * FD * 4);           // current node features
  float* H1  = carve((size_t)NN * FD * 4);           // x @ W
  float* AG  = carve((size_t)NN * FD * 4);           // A @ H (then relu'd conv out)
  float* Am  = carve((size_t)NGRAPH * 256 * 256 * 4);// batched adjacency (64 MB)
  float* em  = carve((size_t)EE * 4);
  float* nm  = carve((size_t)NN * 4);
  float* deg = carve((size_t)NN * 4);
  float* dnv = carve((size_t)NN * 4);
  float* sln = carve((size_t)NN * 4);
  float* sc  = carve((size_t)NN * 4);
  float* z   = carve((size_t)NGRAPH * 256 * 4);
  float* z1  = carve((size_t)NGRAPH * 128 * 4);
  float* z2  = carve((size_t)NGRAPH * 64 * 4);

  hipMemcpyAsync(Xc, x, (size_t)NN * FD * 4, hipMemcpyDeviceToDevice, stream);
  k_fill<<<EE / 256, 256, 0, stream>>>(em, 1.f, EE);
  k_fill<<<NN / 256, 256, 0, stream>>>(nm, 1.f, NN);

  const int kk[3] = {128, 64, 32};
  for (int L = 0; L < 3; ++L) {
    hipMemsetAsync(deg, 0, (size_t)NN * 4, stream);
    hipMemsetAsync(Am, 0, (size_t)NGRAPH * 256 * 256 * 4, stream);
    k_deg<<<EE / 256, 256, 0, stream>>>(dst, em, deg);
    k_dinv<<<NN / 256, 256, 0, stream>>>(deg, nm, dnv);
    k_buildA<<<EE / 256, 256, 0, stream>>>(src, dst, em, dnv, Am);
    k_diag<<<NN / 256, 256, 0, stream>>>(Am, dnv, nm);

    // H1 = Xc @ W  (K=128, shared weight for all blocks)
    k_gemm_wmma<<<512, 256, 128 * 128 * 4, stream>>>(Xc, Wc[L], H1, 128, 0, 1);
    // AG = A_g @ H_g (K=256, 2 blocks per graph share B=H_g)
    k_gemm_wmma<<<512, 256, 256 * 128 * 4, stream>>>(Am, H1, AG, 256, 256 * 128, 2);
    // relu((agg + b) * nmask) -> AG holds conv output
    k_combine<<<NN * FD / 256, 256, 0, stream>>>(AG, bc[L], nm);

    // pooling score conv (H->1) using same A
    k_slin<<<NN / 8, 256, 0, stream>>>(AG, Wp[L], sln);
    k_score<<<NN / 8, 256, 0, stream>>>(Am, sln, bp[L], nm, sc);

    // top-k, apply mask, update features/edges, readout accumulate
    k_topk<<<NGRAPH, 256, 0, stream>>>(sc, nm, kk[L]);
    k_pool<<<NN * FD / 256, 256, 0, stream>>>(AG, sc, nm, Xc);
    k_emask<<<EE / 256, 256, 0, stream>>>(src, dst, nm, em);
    k_readout<<<NGRAPH, 128, 0, stream>>>(Xc, nm, z, L > 0 ? 1 : 0);
  }

  k_mlp1<<<NGRAPH, 128, 0, stream>>>(z, Wl1, bl1, z1);
  k_mlp2<<<NGRAPH, 64, 0, stream>>>(z1, Wl2, bl2, z2);
  k_mlp3<<<NGRAPH, 32, 0, stream>>>(z2, Wl3, bl3, out);
}